// Attention_16741782520069
// MI455X (gfx1250) — compile-verified
//
#include <hip/hip_runtime.h>

// CDNA5 / gfx1250, wave32. Relative-position causal attention.
// One block = one (b,h) pair x 32 query rows. 8 waves of 32 lanes.

typedef __attribute__((ext_vector_type(2))) float v2f;
typedef __attribute__((ext_vector_type(8))) float v8f;

#define SEQ 1024
#define DIM 64
#define QT 32              // query rows per block
#define SSTRIDE 1025       // 1024 + 1 pad: kills 16-way LDS bank conflicts on column reads
#define QSTRIDE 65         // 64 + 1 pad
#define PESTRIDE 66
#define NREL 65            // 2*32+1 relative positions

__global__ __launch_bounds__(256)
void relpos_attn_kernel(const float* __restrict__ Q,
                        const float* __restrict__ K,
                        const float* __restrict__ V,
                        const float* __restrict__ PEk,
                        const float* __restrict__ PEv,
                        float* __restrict__ Out,
                        float* __restrict__ Pout) {
  extern __shared__ float lds[];
  float* sQ   = lds;                          // [QT][QSTRIDE]
  float* sS   = sQ + QT * QSTRIDE;            // [QT][SSTRIDE]  scores -> probs
  float* sQPE = sS + QT * SSTRIDE;            // [QT][PESTRIDE] q . pe[r]

  const int bh   = blockIdx.y;
  const int q0   = blockIdx.x * QT;
  const int tid  = threadIdx.x;
  const int wave = tid >> 5;
  const int ln   = tid & 31;
  const int half = ln >> 4;                   // WMMA lane-group (0: lanes 0-15, 1: 16-31)
  const int m    = ln & 15;

  const float* Qb = Q + (size_t)bh * SEQ * DIM;
  const float* Kb = K + (size_t)bh * SEQ * DIM;
  const float* Vb = V + (size_t)bh * SEQ * DIM;

  // ---- 1. Stage Q tile into LDS (coalesced) ----
  for (int i = tid; i < QT * DIM; i += 256) {
    int r = i >> 6, c = i & 63;
    sQ[r * QSTRIDE + c] = Qb[(size_t)(q0 + r) * DIM + c];
  }
  __syncthreads();

  // ---- 2. qpe[q][r] = dot(Q[q,:], pe_k[r,:])  (65 distinct pe rows only) ----
  for (int i = tid; i < QT * NREL; i += 256) {
    int r = i / NREL, e = i % NREL;
    const float* pr = PEk + e * DIM;
    const float* qr = sQ + r * QSTRIDE;
    float acc = 0.f;
#pragma unroll 8
    for (int d = 0; d < DIM; ++d) acc += qr[d] * pr[d];
    sQPE[r * PESTRIDE + e] = acc;
  }
  __syncthreads();

  // ---- 3. Scores: each wave owns a 128-wide k strip of the 32x1024 tile ----
  // v_wmma_f32_16x16x4_f32:  A 16x4 (lane m = row M; v0:K=0/2, v1:K=1/3 by half)
  //                          B 4x16 (lane m = col N; v0:K=0/2, v1:K=1/3 by half)
  const int kc0 = wave * 128;
#pragma unroll
  for (int qt = 0; qt < 2; ++qt) {
    for (int kt = 0; kt < 8; ++kt) {
      const int kbase = kc0 + kt * 16;
      // prefetch next K tile rows into cache (global_prefetch_b8)
      if (kt < 7)
        __builtin_prefetch(Kb + (size_t)(kbase + 16 + m) * DIM, 0, 1);
      v8f acc = {};
      const float* qrow = sQ + (qt * 16 + m) * QSTRIDE + 2 * half;
      const float* krow = Kb + (size_t)(kbase + m) * DIM + 2 * half;
      for (int dd = 0; dd < DIM; dd += 4) {
        v2f a, b;
        a.x = qrow[dd];     a.y = qrow[dd + 1];
        b.x = krow[dd];     b.y = krow[dd + 1];
        acc = __builtin_amdgcn_wmma_f32_16x16x4_f32(false, a, false, b,
                                                    (short)0, acc, false, false);
      }
      // unpack C layout (VGPR r: rows r / r+8 by half, col = m), fuse bias+scale+mask
#pragma unroll
      for (int r = 0; r < 8; ++r) {
        int row = qt * 16 + r + 8 * half;       // local q row 0..31
        int qg  = q0 + row;
        int kg  = kbase + m;
        int rel = kg - qg;
        rel = rel < -32 ? -32 : (rel > 32 ? 32 : rel);
        float s = (acc[r] + sQPE[row * PESTRIDE + (rel + 32)]) * 0.125f; // 1/sqrt(64)
        if (kg > qg) s = -1e9f;                 // causal mask
        sS[row * SSTRIDE + kg] = s;
      }
    }
  }
  __syncthreads();

  // ---- 4. Row softmax (each wave owns 4 full rows; wave32 shfl reductions) ----
#pragma unroll
  for (int rr = 0; rr < 4; ++rr) {
    const int row = wave * 4 + rr;
    float* srow = sS + row * SSTRIDE;
    float mx = -3.4e38f;
    for (int c = ln; c < SEQ; c += 32) mx = fmaxf(mx, srow[c]);
#pragma unroll
    for (int off = 16; off > 0; off >>= 1) mx = fmaxf(mx, __shfl_xor(mx, off, 32));
    float sum = 0.f;
    for (int c = ln; c < SEQ; c += 32) {
      float e = __expf(srow[c] - mx);
      srow[c] = e;
      sum += e;
    }
#pragma unroll
    for (int off = 16; off > 0; off >>= 1) sum += __shfl_xor(sum, off, 32);
    const float inv = 1.0f / sum;
    for (int c = ln; c < SEQ; c += 32) srow[c] *= inv;
  }
  __syncthreads();

  // ---- 5. Write p_attn tile (coalesced: consecutive tid -> consecutive col) ----
  float* Pb = Pout + (size_t)bh * SEQ * SEQ;
  for (int i = tid; i < QT * SEQ; i += 256) {
    int row = i >> 10, c = i & 1023;
    Pb[(size_t)(q0 + row) * SEQ + c] = sS[row * SSTRIDE + c];
  }

  // ---- 6. P @ V via WMMA: one 16x16 output tile per wave ----
  const int qt = wave >> 2;        // 0..1 : q sub-tile
  const int dt = wave & 3;         // 0..3 : d sub-tile
  v8f oacc = {};
  {
    const float* prow = sS + (qt * 16 + m) * SSTRIDE + 2 * half;
    const float* vcol = Vb + (size_t)(2 * half) * DIM + dt * 16 + m;
    for (int kk = 0; kk < SEQ; kk += 4) {
      v2f a, b;
      a.x = prow[kk];                 a.y = prow[kk + 1];
      b.x = vcol[(size_t)kk * DIM];   b.y = vcol[(size_t)(kk + 1) * DIM];
      oacc = __builtin_amdgcn_wmma_f32_16x16x4_f32(false, a, false, b,
                                                   (short)0, oacc, false, false);
    }
  }

  // ---- 7. Rel-V correction + store output ----
  // out_rel[q,d] = sum_{r=0..32} bucket[r]*pe_v[r,d]; buckets 1..32 are single
  // p elements (k = q-32+r); bucket 0 aggregate = 1 - sum(window) (row sums to 1).
  float* Ob = Out + (size_t)bh * SEQ * DIM;
#pragma unroll
  for (int r = 0; r < 8; ++r) {
    const int row = qt * 16 + r + 8 * half;
    const int qg  = q0 + row;
    const int d   = dt * 16 + m;
    float corr = 0.f, win = 0.f;
#pragma unroll 4
    for (int j = 1; j <= 32; ++j) {
      int kg = qg - 32 + j;
      if (kg >= 0) {
        float pj = sS[row * SSTRIDE + kg];
        corr += pj * PEv[j * DIM + d];
        win  += pj;
      }
    }
    corr += (1.0f - win) * PEv[d];     // pe_v[0][d] * aggregate bucket
    Ob[(size_t)qg * DIM + d] = oacc[r] + corr;
  }
}

extern "C" void kernel_launch(void* const* d_in, const int* in_sizes, int n_in,
                              void* d_out, int out_size, void* d_ws, size_t ws_size,
                              hipStream_t stream) {
  const float* Q   = (const float*)d_in[0];
  const float* K   = (const float*)d_in[1];
  const float* V   = (const float*)d_in[2];
  const float* PEk = (const float*)d_in[3];
  const float* PEv = (const float*)d_in[4];

  const int bh_count = in_sizes[0] / (SEQ * DIM);        // B*H = 64
  float* Out  = (float*)d_out;                           // [B,H,S,D]
  float* Pout = (float*)d_out + (size_t)bh_count * SEQ * DIM;  // [B,H,S,S]

  const size_t shmem = (size_t)(QT * QSTRIDE + QT * SSTRIDE + QT * PESTRIDE) * sizeof(float);
  dim3 grid(SEQ / QT, bh_count);
  relpos_attn_kernel<<<grid, 256, shmem, stream>>>(Q, K, V, PEk, PEv, Out, Pout);
}